// MVPROPFAT_22797686408170
// MI455X (gfx1250) — compile-verified
//
#include <hip/hip_runtime.h>

typedef __attribute__((ext_vector_type(2))) float v2f;
typedef __attribute__((ext_vector_type(8))) float v8f;

#define NPIX   (256 * 256)
#define NBATCH 16
#define NTOT   (NBATCH * NPIX)

// ---------------------------------------------------------------------------
// Setup: repack W_h (150,1,3,3) into WMMA B-fragment layout (4x16 f32 tiles),
// pad bias / W_p to 160 channels.
//   B[k][n] = W_h[n*9 + k]   (lax conv = cross-correlation, no flip)
//   B-frag lane layout (32-bit B, K=4): lane<16 -> {B[k0+0][n], B[k0+1][n]},
//                                       lane>=16 -> {B[k0+2][n], B[k0+3][n]}
// ---------------------------------------------------------------------------
__global__ __launch_bounds__(1024) void mvprop_setup(
    const float* __restrict__ Wh, const float* __restrict__ bh,
    const float* __restrict__ Wp, float2* __restrict__ wB,
    float* __restrict__ wBias, float* __restrict__ wWp) {
  int t = threadIdx.x;
  if (t < 10 * 3 * 32) {                 // 10 N-tiles x 3 K-steps x 32 lanes
    int tile = t / 96;
    int ks   = (t / 32) % 3;
    int lane = t & 31;
    int n    = tile * 16 + (lane & 15);
    int kb   = ks * 4 + ((lane >= 16) ? 2 : 0);
    float b0 = 0.0f, b1 = 0.0f;
    if (n < 150) {
      if (kb     < 9) b0 = Wh[n * 9 + kb];
      if (kb + 1 < 9) b1 = Wh[n * 9 + kb + 1];
    }
    wB[t] = make_float2(b0, b1);
  }
  if (t < 160) {
    wBias[t] = (t < 150) ? bh[t] : 0.0f;
    wWp[t]   = (t < 150) ? Wp[t] : 0.0f;
  }
}

// ---------------------------------------------------------------------------
// Conv stage: p = sigmoid( sum_n Wp[n] * relu( conv3x3(occ)_n + b_n ) )
// One wave handles 16 consecutive pixels x all 150 (pad 160) channels:
// 10 channel tiles x 3 WMMA k-steps = 30 x V_WMMA_F32_16X16X4_F32 per wave.
// ---------------------------------------------------------------------------
__global__ __launch_bounds__(256) void mvprop_conv(
    const float* __restrict__ image, const float2* __restrict__ wB,
    const float* __restrict__ wBias, const float* __restrict__ wWp,
    float* __restrict__ pout) {
  const int lane  = threadIdx.x & 31;
  const int wave  = threadIdx.x >> 5;
  const int tileI = blockIdx.x * 8 + wave;       // 65536 pixel tiles total
  const int base  = tileI * 16;                  // first global pixel of tile
  const int b     = base >> 16;                  // batch
  const int pix   = base & (NPIX - 1);
  const int y     = pix >> 8;
  const int x0    = pix & 255;

  const float* __restrict__ occ = image + (size_t)b * 2 * NPIX;  // channel 0

  const int m     = lane & 15;                   // pixel within tile (M row)
  const int khalf = lane >> 4;                   // 0: k0,k0+1  1: k0+2,k0+3
  const int x     = x0 + m;

  // A fragments: 16x4 f32 im2col patches for k-steps 0..2 (k = 0..11, 9 real)
  v2f aFrag[3];
#pragma unroll
  for (int ks = 0; ks < 3; ++ks) {
    float av[2];
#pragma unroll
    for (int j = 0; j < 2; ++j) {
      int k = ks * 4 + khalf * 2 + j;
      float val = 0.0f;
      if (k < 9) {
        int dy = k / 3 - 1, dx = k % 3 - 1;
        int yy = y + dy, xx = x + dx;
        bool ok = (yy >= 0) & (yy < 256) & (xx >= 0) & (xx < 256);
        int yc = min(max(yy, 0), 255);
        int xc = min(max(xx, 0), 255);
        val = occ[yc * 256 + xc] * (ok ? 1.0f : 0.0f);   // SAME zero-pad
      }
      av[j] = val;
    }
    aFrag[ks].x = av[0];
    aFrag[ks].y = av[1];
  }

  float acc[8];
#pragma unroll
  for (int r = 0; r < 8; ++r) acc[r] = 0.0f;

#pragma unroll
  for (int tile = 0; tile < 10; ++tile) {
    float bias = wBias[tile * 16 + m];           // per-N bias, uniform over M
    v8f c;
#pragma unroll
    for (int r = 0; r < 8; ++r) c[r] = bias;
#pragma unroll
    for (int ks = 0; ks < 3; ++ks) {
      float2 bt = wB[(tile * 3 + ks) * 32 + lane];
      v2f bf;
      bf.x = bt.x;
      bf.y = bt.y;
      // D = A(16x4) x B(4x16) + C  -- v_wmma_f32_16x16x4_f32
      c = __builtin_amdgcn_wmma_f32_16x16x4_f32(
          /*neg_a=*/false, aFrag[ks], /*neg_b=*/false, bf,
          /*c_mod=*/(short)0, c, /*reuse_a=*/false, /*reuse_b=*/false);
    }
    float wp = wWp[tile * 16 + m];               // 1x1 conv weight for this N
#pragma unroll
    for (int r = 0; r < 8; ++r) acc[r] += fmaxf(c[r], 0.0f) * wp;
  }

  // Reduce over N (16 lanes within each half-wave); halves stay separate
  // because xor masks < 16 never flip bit4.
#pragma unroll
  for (int r = 0; r < 8; ++r) {
    float s = acc[r];
    s += __shfl_xor(s, 1, 32);
    s += __shfl_xor(s, 2, 32);
    s += __shfl_xor(s, 4, 32);
    s += __shfl_xor(s, 8, 32);
    acc[r] = s;
  }

  // Lane 0 holds pixels m=0..7, lane 16 holds pixels m=8..15 (contiguous).
  if ((lane & 15) == 0) {
    float sg[8];
#pragma unroll
    for (int r = 0; r < 8; ++r) sg[r] = 1.0f / (1.0f + __expf(-acc[r]));
    float4* dst = (float4*)(pout + base + khalf * 8);
    dst[0] = make_float4(sg[0], sg[1], sg[2], sg[3]);
    dst[1] = make_float4(sg[4], sg[5], sg[6], sg[7]);
  }
}

// ---------------------------------------------------------------------------
// One value-propagation step: v' = max(v, r + p*(maxpool3x3(v) - r)).
// vin is batch-strided so iteration 0 can read r in-place from `image`.
// Final pass fuses the occupancy mask and writes d_out.
// ---------------------------------------------------------------------------
__global__ __launch_bounds__(256) void mvprop_step(
    const float* __restrict__ image, const float* __restrict__ pbuf,
    const float* __restrict__ vin, long vinStride,
    float* __restrict__ vout, int finalPass) {
  int idx = blockIdx.x * 256 + threadIdx.x;
  int b   = idx >> 16;
  int pix = idx & (NPIX - 1);
  int y   = pix >> 8;
  int x   = pix & 255;

  const float* __restrict__ vb = vin + (size_t)b * vinStride;

  float vc = vb[pix];
  float mx = vc;                                 // center always valid
  int y0 = max(y - 1, 0), y1 = min(y + 1, 255);
  int x0 = max(x - 1, 0), x1 = min(x + 1, 255);
  for (int yy = y0; yy <= y1; ++yy)
    for (int xx = x0; xx <= x1; ++xx)
      mx = fmaxf(mx, vb[yy * 256 + xx]);

  const float* ib = image + (size_t)b * 2 * NPIX;
  float r = ib[NPIX + pix];
  float p = pbuf[idx];
  float vn = fmaxf(vc, r + p * (mx - r));

  if (finalPass) {
    float oc = ib[pix];
    vout[idx] = (oc > 0.49f) ? vn : -1.0f;
  } else {
    vout[idx] = vn;
  }
}

// ---------------------------------------------------------------------------
extern "C" void kernel_launch(void* const* d_in, const int* in_sizes, int n_in,
                              void* d_out, int out_size, void* d_ws,
                              size_t ws_size, hipStream_t stream) {
  const float* image = (const float*)d_in[0];   // (16,2,256,256)
  const float* Wh    = (const float*)d_in[1];   // (150,1,3,3)
  const float* bh    = (const float*)d_in[2];   // (150,)
  const float* Wp    = (const float*)d_in[3];   // (1,150,1,1)
  float* out = (float*)d_out;                   // (16,1,256,256)

  float* ws    = (float*)d_ws;
  float* pbuf  = ws;                            // NTOT floats
  float* v0    = ws + (size_t)NTOT;             // NTOT floats
  float* v1    = ws + (size_t)2 * NTOT;         // NTOT floats
  float* wB    = ws + (size_t)3 * NTOT;         // 960 float2 = 1920 floats
  float* wBias = wB + 1920;                     // 160 floats
  float* wWp   = wBias + 160;                   // 160 floats

  mvprop_setup<<<1, 1024, 0, stream>>>(Wh, bh, Wp, (float2*)wB, wBias, wWp);

  mvprop_conv<<<NTOT / (16 * 8), 256, 0, stream>>>(
      image, (const float2*)wB, wBias, wWp, pbuf);

  // K = 20 propagation steps; step 0 reads r directly from image (strided).
  const float* vin = image + NPIX;   // r = image[:,1,:,:]
  long stride = 2 * NPIX;
  float* bufs[2] = {v0, v1};
  for (int i = 0; i < 20; ++i) {
    int fin = (i == 19) ? 1 : 0;
    float* vo = fin ? out : bufs[i & 1];
    mvprop_step<<<NTOT / 256, 256, 0, stream>>>(image, pbuf, vin, stride, vo,
                                                fin);
    vin = vo;
    stride = NPIX;
  }
}